// RelativeSelfAttention_81320910783024
// MI455X (gfx1250) — compile-verified
//
#include <hip/hip_runtime.h>
#include <cstdint>
#include <cstddef>

// ---------------------------------------------------------------------------
// CDNA5 (gfx1250) fp32 WMMA helpers.  D = A(16x4) * B(4x16) + C(16x16), f32.
// ---------------------------------------------------------------------------
typedef __attribute__((ext_vector_type(2))) float v2f;
typedef __attribute__((ext_vector_type(8))) float v8f;

static __device__ __forceinline__ v8f wmma_f32_16x16x4(v2f a, v2f b, v8f c) {
  return __builtin_amdgcn_wmma_f32_16x16x4_f32(false, a, false, b, (short)0, c,
                                               false, false);
}

// ---------------------------------------------------------------------------
// Async global->LDS copy (CDNA5 GLOBAL_LOAD_ASYNC_TO_LDS_B128, ASYNCcnt).
// Probe-confirmed param type: "int __vector_size__(16) __device__ *", i.e.
// non-const AS1 pointer to a 16-byte int vector; LDS side is AS3.
// ---------------------------------------------------------------------------
#if __has_builtin(__builtin_amdgcn_global_load_async_to_lds_b128)
#define HAS_ASYNC_LDS 1
#else
#define HAS_ASYNC_LDS 0
#endif

typedef int i32x4_vs __attribute__((vector_size(4 * sizeof(int))));

static __device__ __forceinline__ void async_copy16(const float* g, float* l) {
#if HAS_ASYNC_LDS
  __builtin_amdgcn_global_load_async_to_lds_b128(
      (__attribute__((address_space(1))) i32x4_vs*)g,
      (__attribute__((address_space(3))) i32x4_vs*)l, 0, 0);
#else
  *(float4*)l = *(const float4*)g;
#endif
}

static __device__ __forceinline__ void async_wait_all() {
#if HAS_ASYNC_LDS
#if __has_builtin(__builtin_amdgcn_s_wait_asynccnt)
  __builtin_amdgcn_s_wait_asynccnt(0);
#else
  asm volatile("s_wait_asynccnt 0x0" ::: "memory");
#endif
#endif
}

// Problem constants
#define BATCH     32
#define CIN       256
#define L         4096          // 64*64 spatial
#define QKV_M     1536          // 3*512
#define HD        512           // per-head "d" extent after reshape
#define SEQ       128           // per-head "m" extent after reshape
#define NHEADS    32
#define HEAD_ELEM 65536         // 512*128 contiguous floats per head tile

// ---------------------------------------------------------------------------
// Generic tiled GEMM:  Out[b, m, n] = sum_k W[m,k] * Xin[b,k,n] + bias[m]
// Block tile 128(m) x 128(n), 8 waves, wave strip = 16 m rows.
// Activation tile DMA'd into double-buffered LDS (2 x 64 x 132) so the async
// copy of chunk i+1 overlaps WMMA work on chunk i.
// ---------------------------------------------------------------------------
__global__ __launch_bounds__(256) void gemm_wmma_kernel(
    const float* __restrict__ W, const float* __restrict__ bias,
    const float* __restrict__ Xin, float* __restrict__ Out,
    int M, int K)
{
  __shared__ __align__(16) float ldsB[2][64][128 + 4];

  const int bz   = blockIdx.z;
  const int m0   = blockIdx.y * 128;
  const int n0   = blockIdx.x * 128;
  const int tid  = threadIdx.x;
  const int wave = tid >> 5;
  const int lane = tid & 31;
  const int l16  = lane & 15;
  const int hi   = lane >> 4;

  const float* Xb = Xin + (size_t)bz * K * L;
  float*       Ob = Out + (size_t)bz * M * L;

  const int arow = m0 + wave * 16 + l16;   // A-fragment row for this lane

  // per-thread copy geometry: 8 b128 items covering the 64x128 tile
  const int cp_k = tid >> 5;               // base k row (0..7) -> +8 per item
  const int cp_n = (tid & 31) << 2;        // n offset within tile

  v8f acc[8] = {};

  // ---- prologue: DMA chunk 0 into buffer 0 ----
  #pragma unroll
  for (int it = 0; it < 8; ++it) {
    const int kk = cp_k + it * 8;
    async_copy16(Xb + (size_t)kk * L + n0 + cp_n, &ldsB[0][kk][cp_n]);
  }
  async_wait_all();
  __syncthreads();

  int buf = 0;
  for (int kt = 0; kt < K; kt += 64) {
    // ---- DMA next chunk into the other buffer while we compute ----
    if (kt + 64 < K) {
      #pragma unroll
      for (int it = 0; it < 8; ++it) {
        const int kk = cp_k + it * 8;
        async_copy16(Xb + (size_t)(kt + 64 + kk) * L + n0 + cp_n,
                     &ldsB[buf ^ 1][kk][cp_n]);
      }
      if (kt + 128 < K) {   // keep L2 warm two chunks ahead
        __builtin_prefetch(Xb + (size_t)(kt + 128 + cp_k) * L + n0 + cp_n, 0, 0);
      }
    }

    // ---- 16 k-steps of 4, 8 column tiles each ----
    #pragma unroll
    for (int kk = 0; kk < 64; kk += 4) {
      v2f a = *(const v2f*)(W + (size_t)arow * K + (kt + kk + hi * 2));
      #pragma unroll
      for (int c = 0; c < 8; ++c) {
        v2f bf;
        bf.x = ldsB[buf][kk + hi * 2][c * 16 + l16];
        bf.y = ldsB[buf][kk + hi * 2 + 1][c * 16 + l16];
        acc[c] = wmma_f32_16x16x4(a, bf, acc[c]);
      }
    }

    async_wait_all();
    __syncthreads();
    buf ^= 1;
  }

  // ---- epilogue: bias add + store (C layout: vgpr r -> row r + 8*hi) ----
  #pragma unroll
  for (int c = 0; c < 8; ++c) {
    #pragma unroll
    for (int r = 0; r < 8; ++r) {
      const int rr = m0 + wave * 16 + r + 8 * hi;
      const int cc = n0 + c * 16 + l16;
      Ob[(size_t)rr * L + cc] = acc[c][r] + bias[rr];
    }
  }
}

// ---------------------------------------------------------------------------
// L2 norm across the HEAD axis (axis=1 of (B,32,512,128)) for q and k.
// For each (b, section t in {q,k}, flat idx in [0,65536)): normalize the 32
// values at stride 65536.  In-place on the qkv buffer.
// ---------------------------------------------------------------------------
__global__ __launch_bounds__(256) void l2norm_heads_kernel(float* __restrict__ qkv)
{
  const size_t tid = (size_t)blockIdx.x * 256 + threadIdx.x; // B*2*65536 threads
  const int idx = (int)(tid & 65535);
  const int t   = (int)((tid >> 16) & 1);       // 0 = q, 1 = k
  const int b   = (int)(tid >> 17);

  float* base = qkv + ((size_t)b * QKV_M + (size_t)t * HD) * L + idx;

  float vals[NHEADS];
  float s = 0.0f;
  #pragma unroll
  for (int h = 0; h < NHEADS; ++h) {
    vals[h] = base[(size_t)h * HEAD_ELEM];
    s += vals[h] * vals[h];
  }
  const float inv = 1.0f / fmaxf(sqrtf(s), 1e-12f);
  #pragma unroll
  for (int h = 0; h < NHEADS; ++h) {
    base[(size_t)h * HEAD_ELEM] = vals[h] * inv;
  }
}

// ---------------------------------------------------------------------------
// Per-(b,h) attention.  Q/K/V head tiles are CONTIGUOUS 512x128 (d x m) blocks.
//   attn[l,m] = sum_d Q[d,l]*K[d,m]   (128x128, K-dim = 512)
//   P = softmax_m(attn)
//   out[d,l]  = sum_m V[d,m]*P[l,m]   (512x128, K-dim = 128)
// Q/K chunks are DMA'd into double-buffered LDS; softmax lives entirely in
// accumulators with half-wave shuffles; P round-trips through LDS for the
// second WMMA GEMM.  Total LDS ~203KB (< 320KB WGP budget).
// ---------------------------------------------------------------------------
__global__ __launch_bounds__(256) void attn_wmma_kernel(
    const float* __restrict__ qkv, float* __restrict__ Obuf)
{
  __shared__ __align__(16) float ldsQ[2][64][132];
  __shared__ __align__(16) float ldsK[2][64][132];
  __shared__ __align__(16) float ldsP[128][132];   // P stored as [l][m]

  const int h    = blockIdx.x;
  const int bz   = blockIdx.y;
  const int tid  = threadIdx.x;
  const int wave = tid >> 5;
  const int lane = tid & 31;
  const int l16  = lane & 15;
  const int hi   = lane >> 4;

  const float* Q  = qkv + (size_t)bz * QKV_M * L + (size_t)h * HEAD_ELEM;
  const float* Kt = Q + (size_t)HD * L;          // + 512*4096
  const float* Vt = Q + (size_t)2 * HD * L;      // + 1024*4096

  // per-thread copy geometry over the 64x128 (8192-float) chunk
  const int cp_d = tid >> 5;                     // base d row, +8 per item
  const int cp_m = (tid & 31) << 2;

  // ---------------- attn = Q^T K ----------------
  v8f acc[8] = {};

  #pragma unroll
  for (int it = 0; it < 8; ++it) {               // prologue DMA, chunk 0
    const int dd = cp_d + it * 8;
    async_copy16(Q  + (size_t)dd * SEQ + cp_m, &ldsQ[0][dd][cp_m]);
    async_copy16(Kt + (size_t)dd * SEQ + cp_m, &ldsK[0][dd][cp_m]);
  }
  async_wait_all();
  __syncthreads();

  int buf = 0;
  for (int d0 = 0; d0 < HD; d0 += 64) {
    if (d0 + 64 < HD) {                          // DMA next chunk
      #pragma unroll
      for (int it = 0; it < 8; ++it) {
        const int dd = cp_d + it * 8;
        async_copy16(Q  + (size_t)(d0 + 64 + dd) * SEQ + cp_m,
                     &ldsQ[buf ^ 1][dd][cp_m]);
        async_copy16(Kt + (size_t)(d0 + 64 + dd) * SEQ + cp_m,
                     &ldsK[buf ^ 1][dd][cp_m]);
      }
    }
    #pragma unroll
    for (int kk = 0; kk < 64; kk += 4) {
      v2f a;                                     // A[l, d] = Q[d, l]
      a.x = ldsQ[buf][kk + hi * 2][wave * 16 + l16];
      a.y = ldsQ[buf][kk + hi * 2 + 1][wave * 16 + l16];
      #pragma unroll
      for (int c = 0; c < 8; ++c) {
        v2f bf;                                  // B[d, m] = K[d, m]
        bf.x = ldsK[buf][kk + hi * 2][c * 16 + l16];
        bf.y = ldsK[buf][kk + hi * 2 + 1][c * 16 + l16];
        acc[c] = wmma_f32_16x16x4(a, bf, acc[c]);
      }
    }
    async_wait_all();
    __syncthreads();
    buf ^= 1;
  }

  // ---------------- softmax over m (128 cols per row) ----------------
  // Row (r + 8*hi) of this wave's strip lives in the 16 lanes of half `hi`
  // across acc[0..7][r]; xor-masks < 16 keep halves independent.
  #pragma unroll
  for (int r = 0; r < 8; ++r) {
    float mx = -3.402823466e38f;
    #pragma unroll
    for (int c = 0; c < 8; ++c) mx = fmaxf(mx, acc[c][r]);
    #pragma unroll
    for (int m = 8; m >= 1; m >>= 1) mx = fmaxf(mx, __shfl_xor(mx, m, 32));
    float sum = 0.0f;
    #pragma unroll
    for (int c = 0; c < 8; ++c) {
      const float e = __expf(acc[c][r] - mx);
      acc[c][r] = e;
      sum += e;
    }
    #pragma unroll
    for (int m = 8; m >= 1; m >>= 1) sum += __shfl_xor(sum, m, 32);
    const float inv = 1.0f / sum;
    const int lrow = wave * 16 + r + 8 * hi;
    #pragma unroll
    for (int c = 0; c < 8; ++c) ldsP[lrow][c * 16 + l16] = acc[c][r] * inv;
  }
  __syncthreads();

  // ---------------- out = V * P^T ----------------
  float* Ob = Obuf + (size_t)bz * HD * L + (size_t)h * HEAD_ELEM;
  #pragma unroll 1
  for (int rt = 0; rt < 4; ++rt) {
    const int drow = wave * 64 + rt * 16 + l16;
    v8f acc2[8] = {};
    #pragma unroll
    for (int k0 = 0; k0 < SEQ; k0 += 4) {
      // A[d, m] = V[d, m]; consecutive m -> contiguous float2 in global
      v2f a = *(const v2f*)(Vt + (size_t)drow * SEQ + k0 + hi * 2);
      #pragma unroll
      for (int c = 0; c < 8; ++c) {
        // B[m, l] = P[l, m] read transposed from ldsP (float2 along m)
        v2f bf = *(const v2f*)&ldsP[c * 16 + l16][k0 + hi * 2];
        acc2[c] = wmma_f32_16x16x4(a, bf, acc2[c]);
      }
    }
    #pragma unroll
    for (int c = 0; c < 8; ++c) {
      #pragma unroll
      for (int r = 0; r < 8; ++r) {
        const int d  = wave * 64 + rt * 16 + r + 8 * hi;
        const int lp = c * 16 + l16;
        Ob[(size_t)d * SEQ + lp] = acc2[c][r];
      }
    }
  }
}

// ---------------------------------------------------------------------------
// Launcher: 4-stage pipeline entirely on `stream`, scratch in d_ws:
//   [0, 768MB)        qkv buffer  (B,1536,4096) f32
//   [768MB, 1024MB+)  attention output (B,512,4096) f32
// ---------------------------------------------------------------------------
extern "C" void kernel_launch(void* const* d_in, const int* in_sizes, int n_in,
                              void* d_out, int out_size, void* d_ws, size_t ws_size,
                              hipStream_t stream) {
  (void)in_sizes; (void)n_in; (void)out_size; (void)ws_size;

  const float* X      = (const float*)d_in[0];
  const float* qkv_w  = (const float*)d_in[1];
  const float* qkv_b  = (const float*)d_in[2];
  const float* proj_w = (const float*)d_in[3];
  const float* proj_b = (const float*)d_in[4];
  float* out = (float*)d_out;

  float* qkv  = (float*)d_ws;
  float* obuf = (float*)((char*)d_ws + (size_t)BATCH * QKV_M * L * sizeof(float));

  // Stage 1: qkv = qkv_w @ X + qkv_b        (M=1536, K=256, N=4096, B=32)
  gemm_wmma_kernel<<<dim3(L / 128, QKV_M / 128, BATCH), 256, 0, stream>>>(
      qkv_w, qkv_b, X, qkv, QKV_M, CIN);

  // Stage 2: l2-normalize q and k across the head axis (in place)
  l2norm_heads_kernel<<<dim3((BATCH * 2 * HEAD_ELEM) / 256), 256, 0, stream>>>(qkv);

  // Stage 3: per-(b,h) attention (softmax(Q^T K) applied to V)
  attn_wmma_kernel<<<dim3(NHEADS, BATCH), 256, 0, stream>>>(qkv, obuf);

  // Stage 4: out = proj_w @ obuf + proj_b   (M=256, K=512, N=4096, B=32)
  gemm_wmma_kernel<<<dim3(L / 128, CIN / 128, BATCH), 256, 0, stream>>>(
      proj_w, proj_b, obuf, out, CIN, HD);
}